// NCDcompReconstructor_1305670058507
// MI455X (gfx1250) — compile-verified
//
#include <hip/hip_runtime.h>

// ---------------------------------------------------------------------------
// Adjoint KB-NUFFT recon for MI455X (gfx1250).
//
//   1) tables kernel: per-sample KB weights (i0 Bessel) + grid indices
//   2) memsetAsync grid planes to zero
//   3) scatter kernel: density-compensated k-space -> oversampled grid
//      (global_atomic_add_f32 into separate R/I planes)
//   4) batched complex GEMM via f32 WMMA with LDS-staged panels
//      (async global->LDS copies via global_load_async_to_lds_b128):
//        stage D: tmp[1280x320] = grid[1280x948] x Ey[948x320]
//        stage E: img[320x320]  = Ex[320x1280]   x tmp[1280x320]
//      948 = 2^2*3*79 -> prime factor makes an FFT ugly; only a 320x320
//      output window survives the crops, so two dense complex DFT matmuls
//      (~58 GFLOP fp32 total) are the natural WMMA formulation, far below
//      both the f32-WMMA compute ceiling and the 23.3 TB/s HBM roofline
//      (total traffic ~0.4 GB; the 146 MB grid fits in the 192 MB L2).
//   5) combine kernel: conj(smaps) coil combine + deapodization + |.|
// ---------------------------------------------------------------------------

#define IM_H   640
#define IM_W   474
#define GX     1280          // 2x oversampled H
#define GY     948           // 2x oversampled W
#define MSAMP  40960
#define NCOILS 15
#define KJ     6             // KB kernel width
#define CROPN  320
#define KB_BETA 14.04f       // 2.34 * J

typedef __attribute__((ext_vector_type(2))) float v2f;
typedef __attribute__((ext_vector_type(8))) float v8f;
typedef int vi4 __attribute__((vector_size(16)));   // builtin's b128 payload type

#if __has_builtin(__builtin_amdgcn_global_load_async_to_lds_b128) && \
    __has_builtin(__builtin_amdgcn_s_wait_asynccnt)
#define USE_ASYNC_LDS 1
#else
#define USE_ASYNC_LDS 0
#endif

static __device__ __forceinline__ v8f wmma_f32(v2f a, v2f b, v8f c) {
    // V_WMMA_F32_16X16X4_F32: D = A(16x4) * B(4x16) + C(16x16)
    return __builtin_amdgcn_wmma_f32_16x16x4_f32(
        /*neg_a=*/false, a, /*neg_b=*/false, b,
        /*c_mod=*/(short)0, c, /*reuse_a=*/false, /*reuse_b=*/false);
}

// 16B global -> LDS copy (zero-fill when !valid, e.g. zero-padded K tail).
static __device__ __forceinline__ void cp_f4(const float* g, float* l, bool valid) {
    if (valid) {
#if USE_ASYNC_LDS
        __builtin_amdgcn_global_load_async_to_lds_b128(
            (vi4*)g, (vi4*)l, /*offset=*/0, /*cpol=*/0);
#else
        *(float4*)l = *(const float4*)g;      // global_load_b128 + ds_store_b128
#endif
    } else {
        *(float4*)l = make_float4(0.f, 0.f, 0.f, 0.f);
    }
}

// Abramowitz & Stegun 9.8.1 / 9.8.2 polynomial approximations of I0(x)
static __device__ __forceinline__ float bessel_i0(float x) {
    float ax = fabsf(x);
    if (ax < 3.75f) {
        float t = x / 3.75f;
        t *= t;
        return 1.0f + t*(3.5156229f + t*(3.0899424f + t*(1.2067492f +
                     t*(0.2659732f + t*(0.0360768f + t*0.0045813f)))));
    } else {
        float t = 3.75f / ax;
        return (expf(ax) * rsqrtf(ax)) *
               (0.39894228f + t*(0.01328592f + t*(0.00225319f +
                t*(-0.00157565f + t*(0.00916281f + t*(-0.02057706f +
                t*(0.02635537f + t*(-0.01647633f + t*0.00392377f))))))));
    }
}

// --------------------------------------------------------------------------
// Kernel 1: per-sample KB interpolation weights + wrapped grid indices
// --------------------------------------------------------------------------
__global__ void kb_tables_kernel(const float* __restrict__ ktraj,  // [2, M]
                                 float* __restrict__ wx, float* __restrict__ wy,
                                 int* __restrict__ ix, int* __restrict__ iy) {
    int m = blockIdx.x * blockDim.x + threadIdx.x;
    if (m >= MSAMP) return;
    const float inv2pi = 0.15915494309189535f;
    float tmx = ktraj[m]          * ((float)GX * inv2pi);
    float tmy = ktraj[MSAMP + m]  * ((float)GY * inv2pi);
    float bx = floorf(tmx - 0.5f * KJ);
    float by = floorf(tmy - 0.5f * KJ);
#pragma unroll
    for (int j = 1; j <= KJ; ++j) {
        float idx = bx + (float)j;
        float u   = tmx - idx;
        float arg = 1.0f - (2.0f * u / KJ) * (2.0f * u / KJ);
        float w   = (arg > 0.0f)
                  ? bessel_i0(KB_BETA * sqrtf(fmaxf(arg, 1e-12f))) : 0.0f;
        wx[m * KJ + j - 1] = w / (float)KJ;
        ix[m * KJ + j - 1] = ((int)idx % GX + GX) % GX;

        idx = by + (float)j;
        u   = tmy - idx;
        arg = 1.0f - (2.0f * u / KJ) * (2.0f * u / KJ);
        w   = (arg > 0.0f)
            ? bessel_i0(KB_BETA * sqrtf(fmaxf(arg, 1e-12f))) : 0.0f;
        wy[m * KJ + j - 1] = w / (float)KJ;
        iy[m * KJ + j - 1] = ((int)idx % GY + GY) % GY;
    }
}

// --------------------------------------------------------------------------
// Kernel 2: DFT twiddle tables; exact integer phase reduction mod G keeps
// float sincos accurate (raw phase would reach ~1000 rad).
//   Ey[ky][ny] = exp(+2*pi*i * ky*(ny-160) / 948)   (948 x 320)
//   Ex[i ][kx] = exp(+2*pi*i * kx*(i -160) / 1280)  (320 x 1280)
// --------------------------------------------------------------------------
__global__ void twiddle_y_kernel(float* __restrict__ eyR, float* __restrict__ eyI) {
    int t = blockIdx.x * blockDim.x + threadIdx.x;
    if (t >= GY * CROPN) return;
    int ky = t / CROPN, ny = t % CROPN;
    long long p = (long long)ky * (long long)(ny - 160);
    int pm = (int)(p % GY); if (pm < 0) pm += GY;
    float ph = 6.283185307179586f * (float)pm / (float)GY;
    float s, c; __sincosf(ph, &s, &c);
    eyR[t] = c; eyI[t] = s;
}

__global__ void twiddle_x_kernel(float* __restrict__ exR, float* __restrict__ exI) {
    int t = blockIdx.x * blockDim.x + threadIdx.x;
    if (t >= CROPN * GX) return;
    int i = t / GX, kx = t % GX;
    long long p = (long long)kx * (long long)(i - 160);
    int pm = (int)(p % GX); if (pm < 0) pm += GX;
    float ph = 6.283185307179586f * (float)pm / (float)GX;
    float s, c; __sincosf(ph, &s, &c);
    exR[t] = c; exI[t] = s;
}

// --------------------------------------------------------------------------
// Kernel 3: gridding scatter. One thread per (sample, 6x6 neighbor), looping
// coils so weight/index are computed once per neighbor.
// --------------------------------------------------------------------------
__global__ void scatter_kernel(const float* __restrict__ ksr,
                               const float* __restrict__ ksi,
                               const float* __restrict__ dcomp,
                               const float* __restrict__ wx,
                               const float* __restrict__ wy,
                               const int*   __restrict__ ix,
                               const int*   __restrict__ iy,
                               float* __restrict__ gridR,
                               float* __restrict__ gridI) {
    int t = blockIdx.x * blockDim.x + threadIdx.x;
    if (t >= MSAMP * KJ * KJ) return;
    int m  = t / (KJ * KJ);
    int jj = t % (KJ * KJ);
    int jx = jj / KJ, jy = jj % KJ;
    float w = wx[m * KJ + jx] * wy[m * KJ + jy];
    int   g = ix[m * KJ + jx] * GY + iy[m * KJ + jy];
    float dc = dcomp[m];
    const size_t plane = (size_t)GX * GY;
#pragma unroll
    for (int c = 0; c < NCOILS; ++c) {
        float yr = ksr[(size_t)c * MSAMP + m] * dc * w;
        float yi = ksi[(size_t)c * MSAMP + m] * dc * w;
        atomicAdd(&gridR[(size_t)c * plane + g], yr);
        atomicAdd(&gridI[(size_t)c * plane + g], yi);
    }
}

// --------------------------------------------------------------------------
// Kernel 4: batched complex GEMM, C[c] = A[c] x B[c]  (row-major planes).
// Block: 256 threads = 8 waves in a 4(x) x 2(y) arrangement; block tile
// 64x32, K staged in chunks of 32 through LDS (zero-padded past Kreal).
//   - A panel 64x32 R/I at LDS stride 36 (16B-aligned float4 staging;
//     36*row mod 64 spreads rows 0..15 over 16 distinct banks for the
//     ds_load_b64 A-fragment reads).
//   - B panel 32x32 R/I at stride 36.
//   - staging via global_load_async_to_lds_b128 + s_wait_asynccnt.
// A-fragment layout for V_WMMA_F32_16X16X4_F32 (16x4 f32): lanes 0-15 hold
// K0,K1 in v[0],v[1]; lanes 16-31 hold K2,K3 -> contiguous 2-float reads.
// Complex product from 4 real WMMAs: D_re = rr - ii, D_im = ri + ir.
// --------------------------------------------------------------------------
#define LDA_S 36
#define LDB_S 36

__global__ __launch_bounds__(256) void zgemm_wmma_kernel(
        const float* __restrict__ AR, const float* __restrict__ AI,
        const float* __restrict__ BR, const float* __restrict__ BI,
        float* __restrict__ CR, float* __restrict__ CI,
        int lda, int ldb, int ldc,
        long strideA, long strideB, long strideC,
        int Kreal, int nChunks, int mTiles, int nTiles) {
    __shared__ __align__(16) float sAR[64 * LDA_S];
    __shared__ __align__(16) float sAI[64 * LDA_S];
    __shared__ __align__(16) float sBR[32 * LDB_S];
    __shared__ __align__(16) float sBI[32 * LDB_S];

    int bid = blockIdx.x;
    int c   = bid / (mTiles * nTiles);
    int rem = bid % (mTiles * nTiles);
    int bm  = rem / nTiles, bn = rem % nTiles;

    const float* aR = AR + (size_t)c * strideA;
    const float* aI = AI + (size_t)c * strideA;
    const float* bR = BR + (size_t)c * strideB;
    const float* bI = BI + (size_t)c * strideB;

    int t = threadIdx.x;
    // staging map: A -> thread (row = t/4, 8 floats at (t%4)*8);
    //              B -> thread (krow = t/8, 4 floats at (t%8)*4)
    int arow  = t >> 2;
    int acol  = (t & 3) * 8;
    int brow  = t >> 3;
    int bcol4 = (t & 7) * 4;

    const float* agR = aR + (size_t)(bm * 64 + arow) * lda;
    const float* agI = aI + (size_t)(bm * 64 + arow) * lda;
    const float* bgR = bR + bn * 32 + bcol4;
    const float* bgI = bI + bn * 32 + bcol4;

    // compute map: 8 waves = 4 (m) x 2 (n) 16x16 complex subtiles
    int lane = t & 31, wid = t >> 5;
    int wtx = wid >> 1, wty = wid & 1;
    int row = lane & 15;          // M row in subtile (and B column)
    int kh  = lane >> 4;          // K half: 0 -> K0,K1 ; 1 -> K2,K3

    v8f rr = {}, ii = {}, ri = {}, ir = {};

    for (int ch = 0; ch < nChunks; ++ch) {
        int k0 = ch * 32;
        __syncthreads();          // previous chunk's LDS reads done
#pragma unroll
        for (int rep = 0; rep < 2; ++rep) {
            int kl = acol + rep * 4;
            int k  = k0 + kl;
            bool v = (k < Kreal);                 // Kreal % 4 == 0
            cp_f4(agR + k, &sAR[arow * LDA_S + kl], v);
            cp_f4(agI + k, &sAI[arow * LDA_S + kl], v);
        }
        {
            int k  = k0 + brow;
            bool v = (k < Kreal);
            cp_f4(bgR + (size_t)k * ldb, &sBR[brow * LDB_S + bcol4], v);
            cp_f4(bgI + (size_t)k * ldb, &sBI[brow * LDB_S + bcol4], v);
        }
#if USE_ASYNC_LDS
        __builtin_amdgcn_s_wait_asynccnt(0);
#else
        __builtin_prefetch(agR + k0 + 32, 0, 0);  // next A chunk into cache
#endif
        __syncthreads();          // panels visible to all waves
#pragma unroll
        for (int kk = 0; kk < 32; kk += 4) {
            v2f aRf = *(const v2f*)&sAR[(wtx * 16 + row) * LDA_S + kk + kh * 2];
            v2f aIf = *(const v2f*)&sAI[(wtx * 16 + row) * LDA_S + kk + kh * 2];
            int kb = kk + kh * 2;
            v2f bRf, bIf;
            bRf.x = sBR[kb * LDB_S + wty * 16 + row];
            bRf.y = sBR[(kb + 1) * LDB_S + wty * 16 + row];
            bIf.x = sBI[kb * LDB_S + wty * 16 + row];
            bIf.y = sBI[(kb + 1) * LDB_S + wty * 16 + row];
            rr = wmma_f32(aRf, bRf, rr);
            ii = wmma_f32(aIf, bIf, ii);
            ri = wmma_f32(aRf, bIf, ri);
            ir = wmma_f32(aIf, bRf, ir);
        }
    }

    float* cRp = CR + (size_t)c * strideC;
    float* cIp = CI + (size_t)c * strideC;
    int gn = bn * 32 + wty * 16 + row;
#pragma unroll
    for (int r = 0; r < 8; ++r) {
        int gm = bm * 64 + wtx * 16 + r + kh * 8;   // C/D layout: VGPR r, lane half
        cRp[(size_t)gm * ldc + gn] = rr[r] - ii[r];
        cIp[(size_t)gm * ldc + gn] = ri[r] + ir[r];
    }
}

// --------------------------------------------------------------------------
// Kernel 5: coil combine with conj(smaps), deapodize, magnitude, final crop.
// --------------------------------------------------------------------------
static __device__ __forceinline__ float deapod_1d(int n, float G) {
    float t = 3.14159265358979f * (float)KJ * (float)n / G;
    float z = KB_BETA * KB_BETA - t * t;
    float sz = sqrtf(fabsf(z) + 1e-12f);
    float K  = (z > 0.0f) ? (sinhf(sz) / sz) : (sinf(sz) / sz);
    return 1.0f / K;
}

__global__ void combine_kernel(const float* __restrict__ imgR,
                               const float* __restrict__ imgI,
                               const float* __restrict__ smR,
                               const float* __restrict__ smI,
                               float* __restrict__ out) {
    int t = blockIdx.x * blockDim.x + threadIdx.x;
    if (t >= CROPN * CROPN) return;
    int i = t / CROPN, j = t % CROPN;
    size_t soff = (size_t)(160 + i) * IM_W + (77 + j);   // pixel in 640x474 image
    float ar = 0.0f, ai = 0.0f;
#pragma unroll
    for (int c = 0; c < NCOILS; ++c) {
        float vr = imgR[(size_t)c * CROPN * CROPN + t];
        float vi = imgI[(size_t)c * CROPN * CROPN + t];
        float sr = smR[(size_t)c * IM_H * IM_W + soff];
        float si = smI[(size_t)c * IM_H * IM_W + soff];
        ar += sr * vr + si * vi;          // conj(s) * v
        ai += sr * vi - si * vr;
    }
    float dea = deapod_1d(i - 160, (float)GX) * deapod_1d(j - 160, (float)GY);
    out[t] = sqrtf(ar * ar + ai * ai) * fabsf(dea);
}

// --------------------------------------------------------------------------
// Host launcher
// --------------------------------------------------------------------------
extern "C" void kernel_launch(void* const* d_in, const int* in_sizes, int n_in,
                              void* d_out, int out_size, void* d_ws, size_t ws_size,
                              hipStream_t stream) {
    (void)in_sizes; (void)n_in; (void)out_size; (void)ws_size;
    const float* ksr   = (const float*)d_in[0];   // [1,15,M]
    const float* ksi   = (const float*)d_in[1];   // [1,15,M]
    const float* ktraj = (const float*)d_in[2];   // [1,2,M]
    const float* dcomp = (const float*)d_in[3];   // [1,M]
    const float* smR   = (const float*)d_in[4];   // [1,15,640,474]
    const float* smI   = (const float*)d_in[5];   // [1,15,640,474]
    float* out = (float*)d_out;                   // [1,320,320]

    // ---- carve workspace (~217 MB) ----
    float* ws = (float*)d_ws;
    size_t o = 0;
    float* wx    = ws + o; o += (size_t)MSAMP * KJ;
    float* wy    = ws + o; o += (size_t)MSAMP * KJ;
    int*   ix    = (int*)(ws + o); o += (size_t)MSAMP * KJ;
    int*   iy    = (int*)(ws + o); o += (size_t)MSAMP * KJ;
    float* eyR   = ws + o; o += (size_t)GY * CROPN;
    float* eyI   = ws + o; o += (size_t)GY * CROPN;
    float* exR   = ws + o; o += (size_t)CROPN * GX;
    float* exI   = ws + o; o += (size_t)CROPN * GX;
    float* gridR = ws + o; o += (size_t)NCOILS * GX * GY;   // contiguous with gridI
    float* gridI = ws + o; o += (size_t)NCOILS * GX * GY;
    float* tmpR  = ws + o; o += (size_t)NCOILS * GX * CROPN;
    float* tmpI  = ws + o; o += (size_t)NCOILS * GX * CROPN;
    float* imgR  = ws + o; o += (size_t)NCOILS * CROPN * CROPN;
    float* imgI  = ws + o; o += (size_t)NCOILS * CROPN * CROPN;

    // ---- 1) KB tables ----
    kb_tables_kernel<<<(MSAMP + 255) / 256, 256, 0, stream>>>(ktraj, wx, wy, ix, iy);

    // ---- 2) twiddles ----
    twiddle_y_kernel<<<(GY * CROPN + 255) / 256, 256, 0, stream>>>(eyR, eyI);
    twiddle_x_kernel<<<(CROPN * GX + 255) / 256, 256, 0, stream>>>(exR, exI);

    // ---- 3) zero both grid planes (contiguous) ----
    (void)hipMemsetAsync(gridR, 0, (size_t)2 * NCOILS * GX * GY * sizeof(float), stream);

    // ---- 4) gridding scatter ----
    const int scat = MSAMP * KJ * KJ;
    scatter_kernel<<<(scat + 255) / 256, 256, 0, stream>>>(
        ksr, ksi, dcomp, wx, wy, ix, iy, gridR, gridI);

    // ---- 5) stage D: tmp = grid x Ey  (M=1280, N=320, K=948 -> pad 960) ----
    {
        int mT = GX / 64, nT = CROPN / 32;                 // 20 x 10
        zgemm_wmma_kernel<<<NCOILS * mT * nT, 256, 0, stream>>>(
            gridR, gridI, eyR, eyI, tmpR, tmpI,
            /*lda=*/GY, /*ldb=*/CROPN, /*ldc=*/CROPN,
            /*strideA=*/(long)GX * GY, /*strideB=*/0L,
            /*strideC=*/(long)GX * CROPN,
            /*Kreal=*/GY, /*nChunks=*/30, mT, nT);
    }

    // ---- 6) stage E: img = Ex x tmp  (M=320, N=320, K=1280) ----
    {
        int mT = CROPN / 64, nT = CROPN / 32;              // 5 x 10
        zgemm_wmma_kernel<<<NCOILS * mT * nT, 256, 0, stream>>>(
            exR, exI, tmpR, tmpI, imgR, imgI,
            /*lda=*/GX, /*ldb=*/CROPN, /*ldc=*/CROPN,
            /*strideA=*/0L, /*strideB=*/(long)GX * CROPN,
            /*strideC=*/(long)CROPN * CROPN,
            /*Kreal=*/GX, /*nChunks=*/40, mT, nT);
    }

    // ---- 7) combine + deapod + magnitude ----
    combine_kernel<<<(CROPN * CROPN + 255) / 256, 256, 0, stream>>>(
        imgR, imgI, smR, smI, out);
}